// LinearIntentMemory_40346922778797
// MI455X (gfx1250) — compile-verified
//
#include <hip/hip_runtime.h>
#include <hip/hip_bf16.h>

// Problem constants (B,T,D) = (4, 2048, 128)
constexpr int Bn = 4;
constexpr int Tn = 2048;
constexpr int Dn = 128;
constexpr int CH = 64;            // chunk length
constexpr int NC = Tn / CH;       // 32 chunks per batch
constexpr int BT = Bn * Tn;       // 8192 tokens

typedef __attribute__((ext_vector_type(2))) float v2f;
typedef __attribute__((ext_vector_type(8))) float v8f;

__device__ __forceinline__ v8f wmma4(v2f a, v2f b, v8f c) {
    // D = A(16x4, f32) * B(4x16, f32) + C(16x16, f32)
    return __builtin_amdgcn_wmma_f32_16x16x4_f32(
        /*neg_a=*/false, a, /*neg_b=*/false, b,
        /*c_mod=*/(short)0, c, /*reuse_a=*/false, /*reuse_b=*/false);
}

// -------------------------------------------------------------------------
// Kernel 1: QKV projection: (8192x128) @ (128x384) + bias, fused elu(x)+1
// on q,k regions. Each wave computes a 16x64 strip (4 WMMA tiles sharing A).
// -------------------------------------------------------------------------
__global__ __launch_bounds__(128) void qkv_kernel(
    const float* __restrict__ x, const float* __restrict__ W,
    const float* __restrict__ bias,
    float* __restrict__ qb, float* __restrict__ kb, float* __restrict__ vb)
{
    const int lane = threadIdx.x & 31;
    const int wave = threadIdx.x >> 5;
    const int lo = lane & 15, hi = lane >> 4;

    const int gw    = blockIdx.x * 4 + wave;  // 3072 waves total
    const int mTile = gw / 6;                 // 512 row strips of 16 tokens
    const int ng    = gw % 6;                 // 6 groups of 64 output cols
    const int row   = mTile * 16 + lo;        // token index (A row)

    v8f acc[4] = {};
    for (int k0 = 0; k0 < Dn; k0 += 4) {
        const int kk = k0 + 2 * hi;
        v2f a; a.x = x[row * Dn + kk]; a.y = x[row * Dn + kk + 1];
#pragma unroll
        for (int tt = 0; tt < 4; ++tt) {
            const int n = ng * 64 + tt * 16 + lo;   // output column 0..383
            // B[kk][n] = W^T[kk][n] = W[n][kk]  (consecutive pair along kk)
            v2f bm; bm.x = W[n * Dn + kk]; bm.y = W[n * Dn + kk + 1];
            acc[tt] = wmma4(a, bm, acc[tt]);
        }
    }

#pragma unroll
    for (int tt = 0; tt < 4; ++tt) {
        const int n = ng * 64 + tt * 16 + lo;
        const float bv = bias[n];
        const int region = n >> 7;            // 0=q, 1=k, 2=v
        const int d = n & 127;
        float* dst = (region == 0) ? qb : (region == 1) ? kb : vb;
#pragma unroll
        for (int r = 0; r < 8; ++r) {
            const int m = mTile * 16 + r + 8 * hi;  // token
            float vv = acc[tt][r] + bv;
            if (region < 2) vv = (vv > 0.0f) ? (vv + 1.0f) : __expf(vv);
            dst[(size_t)m * Dn + d] = vv;
        }
    }
}

// -------------------------------------------------------------------------
// Kernel 2: per-chunk local state S_c = K_c^T V_c (128x128) + local k-sum.
// One block per (batch, chunk); 8 waves, wave w owns rows [16w,16w+16).
// -------------------------------------------------------------------------
__global__ __launch_bounds__(256) void chunk_state_kernel(
    const float* __restrict__ kb, const float* __restrict__ vb,
    float* __restrict__ Sl, float* __restrict__ ksum)
{
    const int bc   = blockIdx.x;              // b*NC + c
    const int tid  = threadIdx.x;
    const int lane = tid & 31, wave = tid >> 5;
    const int lo = lane & 15, hi = lane >> 4;

    const float* kc = kb + (size_t)bc * CH * Dn;
    const float* vc = vb + (size_t)bc * CH * Dn;

    v8f acc[8] = {};
    const int mRow = wave * 16 + lo;          // D-index (row of K^T)
    for (int t0 = 0; t0 < CH; t0 += 4) {
        const int t = t0 + 2 * hi;
        // A[m][t] = K[t][m]
        v2f a; a.x = kc[(size_t)t * Dn + mRow]; a.y = kc[(size_t)(t + 1) * Dn + mRow];
#pragma unroll
        for (int g = 0; g < 8; ++g) {
            const int n = g * 16 + lo;
            v2f bm; bm.x = vc[(size_t)t * Dn + n]; bm.y = vc[(size_t)(t + 1) * Dn + n];
            acc[g] = wmma4(a, bm, acc[g]);
        }
    }

    float* S = Sl + (size_t)bc * Dn * Dn;
#pragma unroll
    for (int g = 0; g < 8; ++g) {
        const int n = g * 16 + lo;
#pragma unroll
        for (int r = 0; r < 8; ++r) {
            const int m = wave * 16 + r + 8 * hi;
            S[(size_t)m * Dn + n] = acc[g][r];
        }
    }

    if (tid < Dn) {
        float s = 0.0f;
        for (int t = 0; t < CH; ++t) s += kc[(size_t)t * Dn + tid];
        ksum[(size_t)bc * Dn + tid] = s;
    }
}

// -------------------------------------------------------------------------
// Kernel 3: exclusive scan over chunks (in place). Each thread owns fixed
// elements; loads all NC values first (independent -> one memory round-trip).
// -------------------------------------------------------------------------
__global__ __launch_bounds__(256) void scan_kernel(
    float* __restrict__ Sl, float* __restrict__ ksum)
{
    const int b = blockIdx.x >> 6;            // 64 blocks per batch
    const int j = blockIdx.x & 63;
    const int e = j * 256 + threadIdx.x;      // 0..16383 element of 128x128

    float* base = Sl + (size_t)b * NC * Dn * Dn + e;
    float vals[NC];
#pragma unroll
    for (int c = 0; c < NC; ++c) vals[c] = base[(size_t)c * Dn * Dn];
    float acc = 0.0f;
#pragma unroll
    for (int c = 0; c < NC; ++c) {
        float t = vals[c];
        base[(size_t)c * Dn * Dn] = acc;      // exclusive prefix
        acc += t;
    }

    if (j == 0 && threadIdx.x < Dn) {
        float* kp = ksum + (size_t)b * NC * Dn + threadIdx.x;
        float kv[NC];
#pragma unroll
        for (int c = 0; c < NC; ++c) kv[c] = kp[(size_t)c * Dn];
        float ka = 0.0f;
#pragma unroll
        for (int c = 0; c < NC; ++c) {
            float t = kv[c];
            kp[(size_t)c * Dn] = ka;
            ka += t;
        }
    }
}

// -------------------------------------------------------------------------
// Kernel 4: per-chunk output.
//   num = causalMask(Q K_c^T) @ V_c + Q @ S_state ; den = rowsum + Q.kstate
// One block per (batch, chunk); 8 waves: wave = (row strip r in 0..3) x
// (column half g in 0..1) of the 64x128 output.
// -------------------------------------------------------------------------
__global__ __launch_bounds__(256) void attn_out_kernel(
    const float* __restrict__ qb, const float* __restrict__ kb,
    const float* __restrict__ vb, const float* __restrict__ Sst,
    const float* __restrict__ kst, float* __restrict__ out)
{
    __shared__ float A_s[CH][CH + 1];         // masked attention scores
    __shared__ float den_s[CH];

    const int bc   = blockIdx.x;              // b*NC + c
    const int tid  = threadIdx.x;
    const int lane = tid & 31, wave = tid >> 5;
    const int lo = lane & 15, hi = lane >> 4;
    const int r4 = wave & 3;                  // row strip (16 timesteps)
    const int g  = wave >> 2;                 // column half (64 cols)

    const float* qc = qb + (size_t)bc * CH * Dn;
    const float* kc = kb + (size_t)bc * CH * Dn;
    const float* vc = vb + (size_t)bc * CH * Dn;
    const float* S  = Sst + (size_t)bc * Dn * Dn;
    const float* kstate = kst + (size_t)bc * Dn;

    const int qRow = r4 * 16 + lo;            // local timestep for A rows

    // Phase A: acc = Q @ S_state (contraction over D=128)
    v8f acc[4] = {};
    for (int i0 = 0; i0 < Dn; i0 += 4) {
        const int i = i0 + 2 * hi;
        v2f a; a.x = qc[(size_t)qRow * Dn + i]; a.y = qc[(size_t)qRow * Dn + i + 1];
#pragma unroll
        for (int tt = 0; tt < 4; ++tt) {
            const int n = g * 64 + tt * 16 + lo;
            v2f bm; bm.x = S[(size_t)i * Dn + n]; bm.y = S[(size_t)(i + 1) * Dn + n];
            acc[tt] = wmma4(a, bm, acc[tt]);
        }
    }

    // Phase B: waves 0..3 compute masked A = tril(Q K^T) into LDS
    if (wave < 4) {
        v8f aacc[4] = {};
        for (int i0 = 0; i0 < Dn; i0 += 4) {
            const int i = i0 + 2 * hi;
            v2f a; a.x = qc[(size_t)qRow * Dn + i]; a.y = qc[(size_t)qRow * Dn + i + 1];
#pragma unroll
            for (int tt = 0; tt < 4; ++tt) {
                const int s = tt * 16 + lo;
                // B[i][s] = K[s][i] (consecutive pair along i)
                v2f bm; bm.x = kc[(size_t)s * Dn + i]; bm.y = kc[(size_t)s * Dn + i + 1];
                aacc[tt] = wmma4(a, bm, aacc[tt]);
            }
        }
#pragma unroll
        for (int tt = 0; tt < 4; ++tt) {
#pragma unroll
            for (int r = 0; r < 8; ++r) {
                const int t = r4 * 16 + r + 8 * hi;
                const int s = tt * 16 + lo;
                A_s[t][s] = (s <= t) ? aacc[tt][r] : 0.0f;
            }
        }
    }
    __syncthreads();

    // Phase C: acc += A_masked @ V (contraction over chunk time, 64)
    for (int s0 = 0; s0 < CH; s0 += 4) {
        const int s = s0 + 2 * hi;
        v2f a; a.x = A_s[r4 * 16 + lo][s]; a.y = A_s[r4 * 16 + lo][s + 1];
#pragma unroll
        for (int tt = 0; tt < 4; ++tt) {
            const int n = g * 64 + tt * 16 + lo;
            v2f bm; bm.x = vc[(size_t)s * Dn + n]; bm.y = vc[(size_t)(s + 1) * Dn + n];
            acc[tt] = wmma4(a, bm, acc[tt]);
        }
    }

    // Phase D: den[t] = rowsum(masked A) + q_t . kstate + 1e-6
    if (tid < CH) {
        float rs = 0.0f;
#pragma unroll
        for (int s = 0; s < CH; ++s) rs += A_s[tid][s];
        float dq = 0.0f;
        for (int i = 0; i < Dn; ++i) dq += qc[(size_t)tid * Dn + i] * kstate[i];
        den_s[tid] = rs + dq + 1e-6f;
    }
    __syncthreads();

    // Phase E: out = num / den
#pragma unroll
    for (int tt = 0; tt < 4; ++tt) {
        const int n = g * 64 + tt * 16 + lo;
#pragma unroll
        for (int r = 0; r < 8; ++r) {
            const int t = r4 * 16 + r + 8 * hi;
            out[(size_t)(bc * CH + t) * Dn + n] = acc[tt][r] / den_s[t];
        }
    }
}

extern "C" void kernel_launch(void* const* d_in, const int* in_sizes, int n_in,
                              void* d_out, int out_size, void* d_ws, size_t ws_size,
                              hipStream_t stream) {
    const float* x = (const float*)d_in[0];   // (B,T,D) fp32
    const float* W = (const float*)d_in[1];   // (3D,D) fp32
    const float* b = (const float*)d_in[2];   // (3D,) fp32
    float* out = (float*)d_out;               // (B,T,D) fp32

    // Workspace layout (floats): q | k | v | chunk states | chunk k-sums
    float* qb = (float*)d_ws;
    float* kb = qb + (size_t)BT * Dn;
    float* vb = kb + (size_t)BT * Dn;
    float* Sl = vb + (size_t)BT * Dn;                       // Bn*NC*Dn*Dn
    float* ks = Sl + (size_t)Bn * NC * Dn * Dn;             // Bn*NC*Dn
    // total ~ 21 MB

    qkv_kernel<<<768, 128, 0, stream>>>(x, W, b, qb, kb, vb);
    chunk_state_kernel<<<Bn * NC, 256, 0, stream>>>(kb, vb, Sl, ks);
    scan_kernel<<<Bn * 64, 256, 0, stream>>>(Sl, ks);
    attn_out_kernel<<<Bn * NC, 256, 0, stream>>>(qb, kb, vb, Sl, ks, out);
}